// LogicGateNetwork_72232759984713
// MI455X (gfx1250) — compile-verified
//
#include <hip/hip_runtime.h>
#include <stdint.h>

// ---------------- Problem constants (from reference) ----------------
#define LGN_BATCH   2048
#define LGN_IN_DIM  8192
#define LGN_HID     16384
#define LGN_OUT     1000
#define LGN_THREADS 512

typedef __attribute__((ext_vector_type(2)))  float        v2f;
typedef __attribute__((ext_vector_type(8)))  float        v8f;
typedef __attribute__((ext_vector_type(4)))  unsigned int v4u;
typedef __attribute__((ext_vector_type(8)))  int          v8i;
typedef __attribute__((ext_vector_type(4)))  int          v4i;

// GATE_COEF (16 gates x 4 basis coefficients [1, a, b, ab])
__device__ __constant__ float kGateCoef[16][4] = {
    {0, 0, 0, 0}, {0, 0, 0, 1}, {0, 1, 0, -1}, {0, 1, 0, 0},
    {0, 0, 1, -1}, {0, 0, 1, 0}, {0, 1, 1, -2}, {0, 1, 1, -1},
    {1, -1, -1, 1}, {1, -1, -1, 2}, {1, 0, -1, 0}, {1, 0, -1, 1},
    {1, -1, 0, 0}, {1, -1, 0, 1}, {1, 0, 0, -1}, {1, 0, 0, 0}};

// Zero-padded B matrix for WMMA: kGateB[k][n] = GATE_COEF[k][n] for n<4 else 0.
// Lets each lane fetch its B element with one computed-address load, no branch.
__device__ __constant__ float kGateB[16][16] = {
    {0, 0, 0, 0,  0,0,0,0,0,0,0,0,0,0,0,0},
    {0, 0, 0, 1,  0,0,0,0,0,0,0,0,0,0,0,0},
    {0, 1, 0, -1, 0,0,0,0,0,0,0,0,0,0,0,0},
    {0, 1, 0, 0,  0,0,0,0,0,0,0,0,0,0,0,0},
    {0, 0, 1, -1, 0,0,0,0,0,0,0,0,0,0,0,0},
    {0, 0, 1, 0,  0,0,0,0,0,0,0,0,0,0,0,0},
    {0, 1, 1, -2, 0,0,0,0,0,0,0,0,0,0,0,0},
    {0, 1, 1, -1, 0,0,0,0,0,0,0,0,0,0,0,0},
    {1, -1, -1, 1, 0,0,0,0,0,0,0,0,0,0,0,0},
    {1, -1, -1, 2, 0,0,0,0,0,0,0,0,0,0,0,0},
    {1, 0, -1, 0, 0,0,0,0,0,0,0,0,0,0,0,0},
    {1, 0, -1, 1, 0,0,0,0,0,0,0,0,0,0,0,0},
    {1, -1, 0, 0, 0,0,0,0,0,0,0,0,0,0,0,0},
    {1, -1, 0, 1, 0,0,0,0,0,0,0,0,0,0,0,0},
    {1, 0, 0, -1, 0,0,0,0,0,0,0,0,0,0,0,0},
    {1, 0, 0, 0,  0,0,0,0,0,0,0,0,0,0,0,0}};

// ====================================================================
// Kernel 1: coef[j][0..3] = softmax(w[j][0..15]) @ GATE_COEF
// WMMA path: one wave handles a 16-neuron tile via 4 chained
// V_WMMA_F32_16X16X4_F32 (K=16 split into 4 chunks of 4).
// A layout (16x4 f32): lanes 0-15 = rows M, VGPR{0,1} = K{0,1} (half 0)
//                      lanes16-31 = rows M, VGPR{0,1} = K{2,3} (half 1)
// B layout (4x16 f32): VGPR v, half h -> row K = v + 2h, lanes = N.
// D layout (16x16 f32): VGPR r, half h -> M = r + 8h, lanes = N.
// All softmax state lives in NAMED scalars (no arrays) so clang cannot
// fold half-wave selects into runtime array indices (-> cndmask chains).
// ====================================================================
__global__ void __launch_bounds__(256)
lgn_coef_kernel(const float* __restrict__ w, float* __restrict__ coef, int out_d)
{
#if defined(__gfx1250__) && __has_builtin(__builtin_amdgcn_wmma_f32_16x16x4_f32)
    const int lane = threadIdx.x & 31;
    const int wave = threadIdx.x >> 5;
    const int tile = blockIdx.x * 8 + wave;   // 16 neurons per tile
    const bool hi  = (lane & 16) != 0;        // half-wave selector
    const int mlow = lane & 15;

    int row  = tile * 16 + mlow;
    int rowc = row < out_d ? row : (out_d - 1);   // clamp loads, keep EXEC all-1s
    const float* wr = w + (size_t)rowc * 16;

    // softmax of w[rowc][0..15] (duplicated across the two half-waves)
    float p0  = wr[0],  p1  = wr[1],  p2  = wr[2],  p3  = wr[3];
    float p4  = wr[4],  p5  = wr[5],  p6  = wr[6],  p7  = wr[7];
    float p8  = wr[8],  p9  = wr[9],  p10 = wr[10], p11 = wr[11];
    float p12 = wr[12], p13 = wr[13], p14 = wr[14], p15 = wr[15];

    const float mx =
        fmaxf(fmaxf(fmaxf(fmaxf(p0, p1),  fmaxf(p2, p3)),
                    fmaxf(fmaxf(p4, p5),  fmaxf(p6, p7))),
              fmaxf(fmaxf(fmaxf(p8, p9),  fmaxf(p10, p11)),
                    fmaxf(fmaxf(p12, p13), fmaxf(p14, p15))));

    p0  = __expf(p0  - mx); p1  = __expf(p1  - mx); p2  = __expf(p2  - mx); p3  = __expf(p3  - mx);
    p4  = __expf(p4  - mx); p5  = __expf(p5  - mx); p6  = __expf(p6  - mx); p7  = __expf(p7  - mx);
    p8  = __expf(p8  - mx); p9  = __expf(p9  - mx); p10 = __expf(p10 - mx); p11 = __expf(p11 - mx);
    p12 = __expf(p12 - mx); p13 = __expf(p13 - mx); p14 = __expf(p14 - mx); p15 = __expf(p15 - mx);

    const float s = (((p0 + p1) + (p2 + p3)) + ((p4 + p5) + (p6 + p7))) +
                    (((p8 + p9) + (p10 + p11)) + ((p12 + p13) + (p14 + p15)));
    const float inv = 1.0f / s;
    p0  *= inv; p1  *= inv; p2  *= inv; p3  *= inv;
    p4  *= inv; p5  *= inv; p6  *= inv; p7  *= inv;
    p8  *= inv; p9  *= inv; p10 *= inv; p11 *= inv;
    p12 *= inv; p13 *= inv; p14 *= inv; p15 *= inv;

    // B fragments: one computed-address constant load per element.
    const int kbofs = hi ? 2 : 0;
    v2f b0, b1, b2, b3;
    b0.x = kGateB[0  + kbofs][mlow]; b0.y = kGateB[1  + kbofs][mlow];
    b1.x = kGateB[4  + kbofs][mlow]; b1.y = kGateB[5  + kbofs][mlow];
    b2.x = kGateB[8  + kbofs][mlow]; b2.y = kGateB[9  + kbofs][mlow];
    b3.x = kGateB[12 + kbofs][mlow]; b3.y = kGateB[13 + kbofs][mlow];

    // A fragments: one cndmask per element (named scalars, no arrays).
    v2f a0; a0.x = hi ? p2  : p0;  a0.y = hi ? p3  : p1;
    v2f a1; a1.x = hi ? p6  : p4;  a1.y = hi ? p7  : p5;
    v2f a2; a2.x = hi ? p10 : p8;  a2.y = hi ? p11 : p9;
    v2f a3; a3.x = hi ? p14 : p12; a3.y = hi ? p15 : p13;

    v8f d = {0.f, 0.f, 0.f, 0.f, 0.f, 0.f, 0.f, 0.f};
    d = __builtin_amdgcn_wmma_f32_16x16x4_f32(false, a0, false, b0, (short)0, d, false, false);
    d = __builtin_amdgcn_wmma_f32_16x16x4_f32(false, a1, false, b1, (short)0, d, false, false);
    d = __builtin_amdgcn_wmma_f32_16x16x4_f32(false, a2, false, b2, (short)0, d, false, false);
    d = __builtin_amdgcn_wmma_f32_16x16x4_f32(false, a3, false, b3, (short)0, d, false, false);

    if (mlow < 4) {   // only N=0..3 columns carry valid coefficients
        #pragma unroll
        for (int r = 0; r < 8; ++r) {
            const int m = r + (hi ? 8 : 0);
            const int neuron = tile * 16 + m;
            if (neuron < out_d) coef[neuron * 4 + mlow] = d[r];
        }
    }
#else
    // Scalar fallback (also what the host pass parses). Same launch geometry.
    const int j = blockIdx.x * blockDim.x + threadIdx.x;
    if (j < out_d) {
        float wv[16];
        float mx = -3.402823466e38f;
        #pragma unroll
        for (int k = 0; k < 16; ++k) { wv[k] = w[j * 16 + k]; mx = fmaxf(mx, wv[k]); }
        float s = 0.f;
        #pragma unroll
        for (int k = 0; k < 16; ++k) { wv[k] = __expf(wv[k] - mx); s += wv[k]; }
        const float inv = 1.0f / s;
        float c0 = 0.f, c1 = 0.f, c2 = 0.f, c3 = 0.f;
        #pragma unroll
        for (int k = 0; k < 16; ++k) {
            const float p = wv[k] * inv;
            c0 = fmaf(p, kGateCoef[k][0], c0);
            c1 = fmaf(p, kGateCoef[k][1], c1);
            c2 = fmaf(p, kGateCoef[k][2], c2);
            c3 = fmaf(p, kGateCoef[k][3], c3);
        }
        coef[j * 4 + 0] = c0; coef[j * 4 + 1] = c1;
        coef[j * 4 + 2] = c2; coef[j * 4 + 3] = c3;
    }
#endif
}

// ====================================================================
// Kernel 2: fully fused 5-layer network. One workgroup = one batch row.
// Activations ping-pong between two 64KB LDS buffers (128KB dynamic LDS
// -> 2 workgroups per 320KB WGP). Indices/coefs are L2/WGP$ resident
// (same addresses for all 2048 blocks).
// ====================================================================
template <int OUT_D>
__device__ __forceinline__ void lgn_layer(const float* __restrict__ in_lds,
                                          float* __restrict__ out_lds,
                                          const int* __restrict__ ia,
                                          const int* __restrict__ ib,
                                          const float4* __restrict__ coef)
{
    // OUT_D is a multiple of LGN_THREADS for hidden layers: exact trip count
    // lets the compiler unroll & software-pipeline the gather->FMA->store chain.
    #pragma unroll 2
    for (int j = threadIdx.x; j < OUT_D; j += LGN_THREADS) {
        const int a_i = ia[j];
        const int b_i = ib[j];
        const float4 c = coef[j];
        const float a = in_lds[a_i];
        const float b = in_lds[b_i];
        // c0 + c1*a + c2*b + c3*a*b  (3 FMAs)
        out_lds[j] = fmaf(a, fmaf(b, c.w, c.y), fmaf(b, c.z, c.x));
    }
}

__global__ void __launch_bounds__(LGN_THREADS)
lgn_fused_kernel(const float* __restrict__ x,
                 const int* __restrict__ ia0, const int* __restrict__ ib0,
                 const int* __restrict__ ia1, const int* __restrict__ ib1,
                 const int* __restrict__ ia2, const int* __restrict__ ib2,
                 const int* __restrict__ ia3, const int* __restrict__ ib3,
                 const int* __restrict__ ia4, const int* __restrict__ ib4,
                 const float* __restrict__ c0, const float* __restrict__ c1,
                 const float* __restrict__ c2, const float* __restrict__ c3,
                 const float* __restrict__ c4,
                 float* __restrict__ out)
{
    extern __shared__ float smem[];
    float* bufA = smem;             // 16384 floats
    float* bufB = smem + LGN_HID;   // 16384 floats

    const int row = blockIdx.x;
    const float* xrow = x + (size_t)row * LGN_IN_DIM;

    // ---- Stage the input row (32KB) into bufA ----
#if defined(__gfx1250__) && __has_builtin(__builtin_amdgcn_tensor_load_to_lds) && __has_builtin(__builtin_amdgcn_s_wait_tensorcnt)
    if (threadIdx.x < 32) {   // one wave issues the TDM DMA
        const uint64_t ga = (uint64_t)(uintptr_t)xrow;
        const unsigned N = LGN_IN_DIM;           // 8192 f32 elements
        v4u g0;
        g0.x = 1u;                                // count=1 (valid descriptor)
        g0.y = 0u;                                // lds_addr: dynamic LDS base
        g0.z = (unsigned)(ga & 0xFFFFFFFFu);      // global_addr[31:0]
        g0.w = ((unsigned)(ga >> 32) & 0x01FFFFFFu) | (2u << 30); // addr[56:32] | type=2
        v8i g1;
        g1[0] = (int)(2u << 16);                  // data_size=4B
        g1[1] = (int)((N & 0xFFFFu) << 16);       // tensor_dim0[15:0]
        g1[2] = (int)((N >> 16) | (1u << 16));    // tensor_dim0[31:16] | tensor_dim1=1
        g1[3] = (int)((N & 0xFFFFu) << 16);       // tensor_dim1 hi=0 | tile_dim0=N
        g1[4] = 1;                                // tile_dim1=1, tile_dim2=0
        g1[5] = (int)N;                           // tensor_dim0_stride lo32
        g1[6] = (int)((N & 0xFFFFu) << 16);       // stride0 hi=0 | stride1[15:0]
        g1[7] = (int)(N >> 16);                   // stride1[47:16]
        v4i g2 = {0, 0, 0, 0};
        v4i g3 = {0, 0, 0, 0};
        v8i g4 = {0, 0, 0, 0, 0, 0, 0, 0};        // extra group (6-arg toolchain form)
        __builtin_amdgcn_tensor_load_to_lds(g0, g1, g2, g3, g4, 0);
        __builtin_amdgcn_s_wait_tensorcnt(0);
    }
    __syncthreads();
#else
    for (int i = threadIdx.x; i < LGN_IN_DIM / 4; i += LGN_THREADS)
        ((float4*)bufA)[i] = ((const float4*)xrow)[i];
    __syncthreads();
#endif

    // ---- 4 hidden logic layers, ping-ponging in LDS ----
    lgn_layer<LGN_HID>(bufA, bufB, ia0, ib0, (const float4*)c0); __syncthreads();
    lgn_layer<LGN_HID>(bufB, bufA, ia1, ib1, (const float4*)c1); __syncthreads();
    lgn_layer<LGN_HID>(bufA, bufB, ia2, ib2, (const float4*)c2); __syncthreads();
    lgn_layer<LGN_HID>(bufB, bufA, ia3, ib3, (const float4*)c3); __syncthreads();

    // ---- Output layer (group=1, TAU=1 => identity reduction), write global ----
    float* orow = out + (size_t)row * LGN_OUT;
    for (int j = threadIdx.x; j < LGN_OUT; j += LGN_THREADS) {
        const int a_i = ia4[j];
        const int b_i = ib4[j];
        const float4 c = ((const float4*)c4)[j];
        const float a = bufA[a_i];
        const float b = bufA[b_i];
        orow[j] = fmaf(a, fmaf(b, c.w, c.y), fmaf(b, c.z, c.x));
    }
}

// ====================================================================
// Launch. Input dict order: x, then (w_i, ia_i, ib_i) for i=0..4.
// ====================================================================
extern "C" void kernel_launch(void* const* d_in, const int* in_sizes, int n_in,
                              void* d_out, int out_size, void* d_ws, size_t ws_size,
                              hipStream_t stream)
{
    (void)in_sizes; (void)n_in; (void)out_size; (void)ws_size;

    const float* x = (const float*)d_in[0];
    const float* w[5];
    const int*   ia[5];
    const int*   ib[5];
    for (int i = 0; i < 5; ++i) {
        w[i]  = (const float*)d_in[1 + 3 * i];
        ia[i] = (const int*)  d_in[2 + 3 * i];
        ib[i] = (const int*)  d_in[3 + 3 * i];
    }

    const int outd[5] = {LGN_HID, LGN_HID, LGN_HID, LGN_HID, LGN_OUT};
    float* coefp[5];
    {
        float* base = (float*)d_ws;
        size_t off = 0;
        for (int i = 0; i < 5; ++i) { coefp[i] = base + off; off += (size_t)outd[i] * 4; }
    }

    // Per-layer coefficient precompute (WMMA): ~1MB total, trivial cost.
    for (int i = 0; i < 5; ++i) {
        const int blocks = (outd[i] + 127) / 128;   // 8 waves x 16 neurons per block
        lgn_coef_kernel<<<blocks, 256, 0, stream>>>(w[i], coefp[i], outd[i]);
    }

    // Fully fused forward pass: one block per batch row, 128KB dynamic LDS.
    const size_t shmem = (size_t)2 * LGN_HID * sizeof(float);
    (void)hipFuncSetAttribute((const void*)lgn_fused_kernel,
                              hipFuncAttributeMaxDynamicSharedMemorySize, (int)shmem);
    lgn_fused_kernel<<<LGN_BATCH, LGN_THREADS, shmem, stream>>>(
        x,
        ia[0], ib[0], ia[1], ib[1], ia[2], ib[2], ia[3], ib[3], ia[4], ib[4],
        coefp[0], coefp[1], coefp[2], coefp[3], coefp[4],
        (float*)d_out);
}